// End2EndPoseLoss_85796266704954
// MI455X (gfx1250) — compile-verified
//
#include <hip/hip_runtime.h>
#include <math.h>

// Problem geometry (from reference): B=16, P=20, K=17, H=W=64
#define NB 16
#define NP 20
#define KHW 69632            // 17*64*64 elements per (b,p) slice
#define SLICES 320           // B*P
#define BLOCKS_PER_SLICE 17
#define F4_PER_BLOCK 1024    // (KHW/4) / BLOCKS_PER_SLICE
#define ITERS 4              // F4_PER_BLOCK / 256
#define NBLOCKS (SLICES * BLOCKS_PER_SLICE)   // 5440

typedef __attribute__((ext_vector_type(2))) float v2f;
typedef __attribute__((ext_vector_type(8))) float v8f;

// Wave32 full-wave sum using the matrix unit.
// A: 16x4 f32, lane L<16 holds (m=L,k=0),(m=L,k=1) in its 2 VGPRs,
//    lane L>=16 holds (m=L-16,k=2),(m=L-16,k=3).
// With a.y=0 and B=ones: D[m,n] = partial[m] + partial[m+16] for all n.
// D layout: lane n<16: d[r]=D[r,n]; lane n>=16: d[r]=D[r+8,n-16].
// So sum(d[0..7]) gives lanes 0-15 the low-half row-sum, lanes 16-31 the
// high half; one xor-16 shuffle completes the 32-lane total in every lane.
__device__ __forceinline__ float wave_sum_wmma(float v) {
    v2f a; a.x = v;   a.y = 0.0f;
    v2f b; b.x = 1.0f; b.y = 1.0f;
    v8f c = {};
    v8f d = __builtin_amdgcn_wmma_f32_16x16x4_f32(
        /*neg_a=*/false, a, /*neg_b=*/false, b,
        /*c_mod=*/(short)0, c, /*reuse_a=*/false, /*reuse_b=*/false);
    float t = ((d[0] + d[1]) + (d[2] + d[3])) + ((d[4] + d[5]) + (d[6] + d[7]));
    t += __shfl_xor(t, 16, 32);
    return t;
}

__device__ __forceinline__ float peak_term(float p, float g) {
    float df = p - g;
    float w = (g > 0.2f) ? 5.0f : 1.0f;
    return df * df * w;
}

// Pass 1: per-block weighted-SSE partials over the big heatmaps.
// mask is uniform per block -> masked-out slices do zero memory traffic.
__global__ __launch_bounds__(256) void heat_partial_kernel(
    const float* __restrict__ pred, const float* __restrict__ gt,
    const int* __restrict__ mask, float* __restrict__ partials) {
    const int blk   = blockIdx.x;
    const int slice = blk / BLOCKS_PER_SLICE;
    const int sub   = blk % BLOCKS_PER_SLICE;

    float acc = 0.0f;
    if (mask[slice] != 0) {
        const float4* p4 = (const float4*)(pred + (size_t)slice * KHW);
        const float4* g4 = (const float4*)(gt   + (size_t)slice * KHW);
        const int base = sub * F4_PER_BLOCK + threadIdx.x;
#pragma unroll
        for (int it = 0; it < ITERS; ++it) {
            float4 p = p4[base + it * 256];
            float4 g = g4[base + it * 256];
            acc += peak_term(p.x, g.x);
            acc += peak_term(p.y, g.y);
            acc += peak_term(p.z, g.z);
            acc += peak_term(p.w, g.w);
        }
    }

    // wave32 reduction via v_wmma_f32_16x16x4_f32 (uniform control flow; EXEC all 1s)
    float ws = wave_sum_wmma(acc);

    __shared__ float sdata[8];
    const int lane = threadIdx.x & 31;
    const int wave = threadIdx.x >> 5;
    if (lane == 0) sdata[wave] = ws;
    __syncthreads();
    if (threadIdx.x == 0) {
        float s = 0.0f;
#pragma unroll
        for (int i = 0; i < 8; ++i) s += sdata[i];
        partials[blk] = s;   // always written -> no stale workspace state
    }
}

// Pass 2: single block. Deterministic tree reductions + the tiny loss terms.
__global__ __launch_bounds__(256) void finalize_kernel(
    const float* __restrict__ partials,
    const float* __restrict__ count_logits,   // (16,21)
    const int*   __restrict__ count,          // (16,)
    const float* __restrict__ conf_logits,    // (320,)
    const int*   __restrict__ mask,           // (320,)
    float* __restrict__ out) {
    __shared__ float rH[256];
    __shared__ float rM[256];
    __shared__ float rF[256];
    __shared__ float ce[16];
    const int tid = threadIdx.x;

    // heatmap partials (fixed order per thread -> deterministic)
    float hs = 0.0f;
    for (int i = tid; i < NBLOCKS; i += 256) hs += partials[i];

    // mask sum + focal loss over 320 conf logits
    float ms = 0.0f, fs = 0.0f;
    for (int i = tid; i < SLICES; i += 256) {
        float t = (float)mask[i];
        ms += t;
        float l   = conf_logits[i];
        float bce = fmaxf(l, 0.0f) - l * t + log1pf(expf(-fabsf(l)));
        float pt  = expf(-bce);
        float om  = 1.0f - pt;
        fs += om * om * bce;
    }
    rH[tid] = hs; rM[tid] = ms; rF[tid] = fs;

    // count cross-entropy: one row (21 logits) per lane for tid<16
    if (tid < 16) {
        const float* row = count_logits + tid * (NP + 1);
        float mx = row[0];
        for (int j = 1; j <= NP; ++j) mx = fmaxf(mx, row[j]);
        float se = 0.0f;
        for (int j = 0; j <= NP; ++j) se += expf(row[j] - mx);
        float lse = mx + logf(se);
        ce[tid] = lse - row[count[tid]];
    }
    __syncthreads();

    for (int s = 128; s > 0; s >>= 1) {
        if (tid < s) {
            rH[tid] += rH[tid + s];
            rM[tid] += rM[tid + s];
            rF[tid] += rF[tid + s];
        }
        __syncthreads();
    }

    if (tid == 0) {
        float cesum = 0.0f;
#pragma unroll
        for (int i = 0; i < 16; ++i) cesum += ce[i];
        float loss_count = cesum / (float)NB;

        float masksum   = rM[0];
        float hm        = rH[0] / (masksum * (float)KHW + 1e-6f);
        float loss_heat = (masksum > 0.0f) ? hm : 0.0f;

        float loss_conf = rF[0] / (float)SLICES;

        out[0] = 1.0f * loss_count + 10.0f * loss_heat + 1.5f * loss_conf;
    }
}

extern "C" void kernel_launch(void* const* d_in, const int* in_sizes, int n_in,
                              void* d_out, int out_size, void* d_ws, size_t ws_size,
                              hipStream_t stream) {
    (void)in_sizes; (void)n_in; (void)out_size; (void)ws_size;
    const float* count_logits = (const float*)d_in[0];  // (16,21)
    const float* pred         = (const float*)d_in[1];  // (16,20,17,64,64)
    const float* conf_logits  = (const float*)d_in[2];  // (16,20)
    const float* gt           = (const float*)d_in[3];  // (16,20,17,64,64)
    const int*   count        = (const int*)d_in[4];    // (16,)
    const int*   mask         = (const int*)d_in[5];    // (16,20)

    float* partials = (float*)d_ws;                     // 5440 floats

    heat_partial_kernel<<<NBLOCKS, 256, 0, stream>>>(pred, gt, mask, partials);
    finalize_kernel<<<1, 256, 0, stream>>>(partials, count_logits, count,
                                           conf_logits, mask, (float*)d_out);
}